// TwoWordNNLabelProbe_4148938408321
// MI455X (gfx1250) — compile-verified
//
#include <hip/hip_runtime.h>
#include <hip/hip_bf16.h>

typedef __attribute__((ext_vector_type(2))) float v2f;
typedef __attribute__((ext_vector_type(8))) float v8f;

#define DDIM 1024
#define RANK 256
#define NB   16
#define SS   256
#define NROWS (NB * SS)    // 4096
#define NCOLS (2 * RANK)   // 512

// ---------------------------------------------------------------------------
// Kernel 1: C[s, n] = sum_d batch[s,d] * Wsel[n,d]   (n<256 -> left + b1,
//                                                     n>=256 -> right)
// One wave computes a 32x64 tile via 2x4 WMMA f32 16x16x4 accumulators.
// ---------------------------------------------------------------------------
__global__ __launch_bounds__(32) void gemm_lr_kernel(
    const float* __restrict__ batch, const float* __restrict__ W1,
    const float* __restrict__ b1, float* __restrict__ leftp,
    float* __restrict__ rightp)
{
  const int lane = threadIdx.x;            // 0..31
  const int m0   = blockIdx.x * 32;        // row tile (s)
  const int n0   = blockIdx.y * 64;        // col tile (r / r+256)
  const int mrow = lane & 15;
  const int kb   = (lane >> 4) * 2;        // 0 or 2 (K sub-offset per lane half)

  // A fragment base word offsets (contiguous float2 per lane)
  const long aBase0 = (long)(m0 + mrow) * DDIM + kb;
  const long aBase1 = (long)(m0 + 16 + mrow) * DDIM + kb;

  // B fragment base word offsets: B[k,n] = W1[row(n), d(k)]
  long wBase[4];
#pragma unroll
  for (int t = 0; t < 4; ++t) {
    int ncol = n0 + t * 16 + (lane & 15);
    int side = (ncol >= RANK) ? 1 : 0;     // 0 = left half, 1 = right half
    int wrow = ncol - side * RANK;
    wBase[t] = (long)wrow * (2 * DDIM) + side * DDIM + kb;
  }

  v8f acc[2][4];
#pragma unroll
  for (int i = 0; i < 2; ++i)
#pragma unroll
    for (int t = 0; t < 4; ++t) acc[i][t] = (v8f){0.f,0.f,0.f,0.f,0.f,0.f,0.f,0.f};

#pragma unroll 2
  for (int k0 = 0; k0 < DDIM; k0 += 4) {
    v2f a0 = *(const v2f*)(batch + aBase0 + k0);
    v2f a1 = *(const v2f*)(batch + aBase1 + k0);
    v2f bf[4];
#pragma unroll
    for (int t = 0; t < 4; ++t) bf[t] = *(const v2f*)(W1 + wBase[t] + k0);
#pragma unroll
    for (int t = 0; t < 4; ++t) {
      acc[0][t] = __builtin_amdgcn_wmma_f32_16x16x4_f32(
          false, a0, false, bf[t], (short)0, acc[0][t], false, false);
      acc[1][t] = __builtin_amdgcn_wmma_f32_16x16x4_f32(
          false, a1, false, bf[t], (short)0, acc[1][t], false, false);
    }
  }

  // Epilogue: scatter C (+ b1 on the left half)
  const int rowoff = (lane >> 4) ? 8 : 0;
#pragma unroll
  for (int t = 0; t < 4; ++t) {
    int ncol = n0 + t * 16 + (lane & 15);
    int side = (ncol >= RANK) ? 1 : 0;
    int ccol = ncol - side * RANK;
    float bias = side ? 0.f : b1[ncol];
    float* dst = side ? rightp : leftp;
#pragma unroll
    for (int i = 0; i < 2; ++i) {
#pragma unroll
      for (int v = 0; v < 8; ++v) {
        int row = m0 + i * 16 + rowoff + v;
        dst[(long)row * RANK + ccol] = acc[i][t][v] + bias;
      }
    }
  }
}

// ---------------------------------------------------------------------------
// Kernel 2: logits[b,i,j] = sum_r relu(left[b,i,r] + right[b,j,r]) * w[r] + bl
// 256 threads / block, 64x64 output tile, 4x4 per thread, LDS-staged rows
// with pitch 257 (bank-conflict-free strided column reads).
// ---------------------------------------------------------------------------
#define PITCH 257
__global__ __launch_bounds__(256) void pair_relu_dot_kernel(
    const float* __restrict__ leftp, const float* __restrict__ rightp,
    const float* __restrict__ wlast, const float* __restrict__ blast,
    float* __restrict__ out)
{
  __shared__ float Ls[64][PITCH];
  __shared__ float Rs[64][PITCH];
  __shared__ float ws[RANK];

  const int t  = threadIdx.x;           // 0..255
  const int jb = blockIdx.x;            // 0..3
  const int ib = blockIdx.y;            // 0..3
  const int bz = blockIdx.z;            // 0..15
  const int si = bz * SS + ib * 64;
  const int sj = bz * SS + jb * 64;

  // Stage 64 rows x 256 cols of left and right into LDS.
  {
    int row = t & 63;                   // lanes in a wave -> distinct rows
    int q   = t >> 6;                   // 0..3 column quarter
    const float4* lr = (const float4*)(leftp  + (long)(si + row) * RANK);
    const float4* rr = (const float4*)(rightp + (long)(sj + row) * RANK);
#pragma unroll
    for (int u = 0; u < 16; ++u) {
      float4 lv = lr[q * 16 + u];
      float4 rv = rr[q * 16 + u];
      int c = q * 64 + u * 4;
      Ls[row][c]   = lv.x; Ls[row][c+1] = lv.y;
      Ls[row][c+2] = lv.z; Ls[row][c+3] = lv.w;
      Rs[row][c]   = rv.x; Rs[row][c+1] = rv.y;
      Rs[row][c+2] = rv.z; Rs[row][c+3] = rv.w;
    }
    ws[t] = wlast[t];
  }
  __syncthreads();

  const int ti = t >> 4;                // 0..15
  const int tj = t & 15;                // 0..15
  float acc[4][4] = {{0.f}};

#pragma unroll 2
  for (int r = 0; r < RANK; ++r) {
    float la[4], rb[4];
#pragma unroll
    for (int a = 0; a < 4; ++a) la[a] = Ls[ti * 4 + a][r];
#pragma unroll
    for (int b = 0; b < 4; ++b) rb[b] = Rs[tj * 4 + b][r];
    float w = ws[r];
#pragma unroll
    for (int a = 0; a < 4; ++a)
#pragma unroll
      for (int b = 0; b < 4; ++b) {
        float h = fmaxf(la[a] + rb[b], 0.f);
        acc[a][b] = __builtin_fmaf(h, w, acc[a][b]);
      }
  }

  const float bl = blast[0];
#pragma unroll
  for (int a = 0; a < 4; ++a) {
#pragma unroll
    for (int b = 0; b < 4; ++b) {
      int i = ib * 64 + ti * 4 + a;
      int j = jb * 64 + tj * 4 + b;
      out[(long)bz * (SS * SS) + (long)i * SS + j] = acc[a][b] + bl;
    }
  }
}

extern "C" void kernel_launch(void* const* d_in, const int* in_sizes, int n_in,
                              void* d_out, int out_size, void* d_ws, size_t ws_size,
                              hipStream_t stream) {
  (void)in_sizes; (void)n_in; (void)out_size; (void)ws_size;
  const float* batch  = (const float*)d_in[0];   // (16,256,1024) f32
  const float* W1     = (const float*)d_in[1];   // (256,2048)    f32
  const float* b1     = (const float*)d_in[2];   // (256,)        f32
  const float* W_last = (const float*)d_in[3];   // (1,256)       f32
  const float* b_last = (const float*)d_in[4];   // (1,)          f32
  float* out = (float*)d_out;                    // (16,256,256)  f32

  float* leftp  = (float*)d_ws;                          // 4096*256 f32
  float* rightp = leftp + (size_t)NROWS * RANK;          // 4096*256 f32 (8 MB total)

  dim3 g1(NROWS / 32, NCOLS / 64);   // (128, 8), 1 wave per block
  gemm_lr_kernel<<<g1, 32, 0, stream>>>(batch, W1, b1, leftp, rightp);

  dim3 g2(4, 4, NB);                 // 256 blocks x 256 threads
  pair_relu_dot_kernel<<<g2, 256, 0, stream>>>(leftp, rightp, W_last, b_last, out);
}